// Loss_60567628808292
// MI455X (gfx1250) — compile-verified
//
#include <hip/hip_runtime.h>

typedef float v2f __attribute__((ext_vector_type(2)));
typedef float v8f __attribute__((ext_vector_type(8)));

#define OBJ_LIMIT 49              // S*S
#define CHUNK 1024                // cells scanned per iteration in cutoff search
#define BLK 256

// ---------------------------------------------------------------------------
// Kernel 1: find P = global index of the 49th object cell (sentinel M-1 if
// fewer than 49 object cells exist). Single block, ordered chunk scan with
// early exit — typically terminates after the first 1024 cells.
// ---------------------------------------------------------------------------
__global__ void find_cutoff(const float* __restrict__ tgt, int* __restrict__ Pptr, int M) {
    __shared__ int scan[BLK];
    __shared__ int s_run, s_P, s_done;
    const int t = threadIdx.x;
    if (t == 0) { s_run = 0; s_P = M - 1; s_done = 0; }
    __syncthreads();

    for (int base = 0; base < M; base += CHUNK) {
        const int idx0 = base + t * 4;           // 4 consecutive cells per thread
        int o[4];
        int cnt = 0;
#pragma unroll
        for (int k = 0; k < 4; ++k) {
            const int idx = idx0 + k;
            const bool b = (idx < M) && (tgt[(size_t)idx * 30 + 4] > 0.0f);
            o[k] = b ? 1 : 0;
            cnt += o[k];
        }
        // speculative prefetch of next chunk's conf channel (gfx1250 global_prefetch)
        if (base + CHUNK < M)
            __builtin_prefetch(&tgt[(size_t)(idx0 + CHUNK) * 30 + 4], 0, 0);

        // inclusive Hillis-Steele scan of per-thread counts
        scan[t] = cnt;
        __syncthreads();
        for (int off = 1; off < BLK; off <<= 1) {
            const int v = (t >= off) ? scan[t - off] : 0;
            __syncthreads();
            scan[t] += v;
            __syncthreads();
        }
        const int incl  = scan[t];
        const int excl  = incl - cnt;
        const int total = scan[BLK - 1];
        const int run   = s_run;

        int r = run + excl;
#pragma unroll
        for (int k = 0; k < 4; ++k) {
            if (o[k]) {
                ++r;
                if (r == OBJ_LIMIT) { s_P = idx0 + k; s_done = 1; }
            }
        }
        __syncthreads();
        if (t == 0) s_run = run + total;
        __syncthreads();
        if (s_done) break;
    }
    if (t == 0) *Pptr = s_P;
}

// ---------------------------------------------------------------------------
// per-box terms (only executed for <=49 active cells total)
// ---------------------------------------------------------------------------
__device__ __forceinline__ void box_terms(const float* bp, const float* bt,
                                          float& tot, float& iou) {
    // xy = box[:2]/7 - 0.5*wh ; x2y2 = xy/7 + 0.5*wh  (reference's buggy transform)
    const float pxx = bp[0] / 7.0f - 0.5f * bp[2];
    const float pxy = bp[1] / 7.0f - 0.5f * bp[3];
    const float p2x = pxx / 7.0f + 0.5f * bp[2];
    const float p2y = pxy / 7.0f + 0.5f * bp[3];
    const float txx = bt[0] / 7.0f - 0.5f * bt[2];
    const float tyy = bt[1] / 7.0f - 0.5f * bt[3];
    const float t2x = txx / 7.0f + 0.5f * bt[2];
    const float t2y = tyy / 7.0f + 0.5f * bt[3];

    const float dx = txx - pxx, dy = tyy - pxy;
    const float l1 = 5.0f * dx * dx + dy * dy;            // lambda_coord on x only
    const float sx = sqrtf(t2x) - sqrtf(p2x);
    const float sy = sqrtf(t2y) - sqrtf(p2y);
    const float l2 = 5.0f * sx * sx + sy * sy;            // lambda_coord on w only
    const float dc = bt[4] - bp[4];
    const float l3 = dc * dc;

    const float ltx = fmaxf(pxx, txx), lty = fmaxf(pxy, tyy);
    const float rbx = fminf(p2x, t2x), rby = fminf(p2y, t2y);
    const float wx = fmaxf(rbx - ltx, 0.0f), wy = fmaxf(rby - lty, 0.0f);
    const float inter = wx * wy;
    const float ap = (p2x - pxx) * (p2y - pxy);
    const float at = (t2x - txx) * (t2y - tyy);
    iou = inter / (ap + at - inter);
    tot = l1 + l2 + l3 + iou;                              // IoU added (reference bug)
}

// ---------------------------------------------------------------------------
// Kernel 2: streaming main pass. One cell per thread; only the first 40 bytes
// of each 120-byte row are needed -> five 8-byte non-temporal loads per tensor.
// ---------------------------------------------------------------------------
__global__ void yolo_main(const float* __restrict__ pred, const float* __restrict__ tgt,
                          const int* __restrict__ Pptr, float* __restrict__ partials,
                          int M) {
    const int P = *Pptr;                    // uniform scalar
    const int i = blockIdx.x * BLK + threadIdx.x;
    float acc = 0.0f;

    if (i < M) {
        const float* pr = pred + (size_t)i * 30;
        const float* tr = tgt  + (size_t)i * 30;
        float p[10], t[10];
#pragma unroll
        for (int k = 0; k < 5; ++k) {       // rows are 8B-aligned (120*i)
            const v2f vp = __builtin_nontemporal_load((const v2f*)pr + k);
            const v2f vt = __builtin_nontemporal_load((const v2f*)tr + k);
            p[2 * k] = vp.x; p[2 * k + 1] = vp.y;
            t[2 * k] = vt.x; t[2 * k + 1] = vt.y;
        }
        const bool obj = t[4] > 0.0f;
        if (!obj) {
            const float d0 = p[4] - t[4];
            const float d1 = p[9] - t[9];
            acc = 0.5f * (d0 * d0 + d1 * d1);            // L_NOOBJ = 0.5
        } else if (i <= P) {
            float tot0, iou0, tot1, iou1;
            box_terms(p,     t,     tot0, iou0);
            box_terms(p + 5, t + 5, tot1, iou1);
            acc = (iou0 >= iou1) ? tot0 : tot1;          // argmax picks first on tie
        }
    }

    __shared__ float red[BLK];
    red[threadIdx.x] = acc;
    __syncthreads();
    for (int off = BLK / 2; off > 0; off >>= 1) {
        if (threadIdx.x < off) red[threadIdx.x] += red[threadIdx.x + off];
        __syncthreads();
    }
    if (threadIdx.x == 0) partials[blockIdx.x] = red[0];
}

// ---------------------------------------------------------------------------
// Kernel 3: deterministic final reduction; last 64->1 stage uses
// v_wmma_f32_16x16x4_f32 (A = 64 partials, B = ones => D rows are row-sums).
// ---------------------------------------------------------------------------
__global__ void final_reduce(const float* __restrict__ partials, int n,
                             float* __restrict__ out) {
    __shared__ float red[BLK];
    const int t = threadIdx.x;
    float local = 0.0f;
    for (int i = t; i < n; i += BLK) local += partials[i];   // fixed order per thread
    red[t] = local;
    __syncthreads();
    if (t < 128) red[t] += red[t + 128];
    __syncthreads();
    if (t < 64)  red[t] += red[t + 64];
    __syncthreads();

    if (t < 32) {                         // full wave -> EXEC all ones for WMMA
        // A-matrix 16x4 f32 layout: lanes 0-15 hold K0,K1 of row m=lane;
        // lanes 16-31 hold K2,K3 of row m=lane-16. Value v = red[4*m + k].
        const int m  = t & 15;
        const int kb = (t >= 16) ? 2 : 0;
        v2f a; a.x = red[4 * m + kb]; a.y = red[4 * m + kb + 1];
        v2f b; b.x = 1.0f; b.y = 1.0f;    // B = ones => D[m,n] = sum_k A[m,k]
        v8f c = {};
        v8f d = __builtin_amdgcn_wmma_f32_16x16x4_f32(
            false, a, false, b, (short)0, c, false, false);
        float s = d[0] + d[1] + d[2] + d[3] + d[4] + d[5] + d[6] + d[7];
        // lanes 0-15 hold sum(rows 0-7), lanes 16-31 hold sum(rows 8-15)
        const float other = __shfl(s, t ^ 16, 32);
        if (t == 0) out[0] = s + other;
    }
}

// ---------------------------------------------------------------------------
extern "C" void kernel_launch(void* const* d_in, const int* in_sizes, int n_in,
                              void* d_out, int out_size, void* d_ws, size_t ws_size,
                              hipStream_t stream) {
    (void)n_in; (void)out_size; (void)ws_size;
    const float* pred = (const float*)d_in[0];
    const float* tgt  = (const float*)d_in[1];
    float* out = (float*)d_out;

    const int M = in_sizes[0] / 30;                 // 802,816 cells
    const int nblocks = (M + BLK - 1) / BLK;        // 3,136 blocks

    float* partials = (float*)d_ws;                 // [0, nblocks) floats
    int*   Pptr     = (int*)d_ws + nblocks;         // one int after partials

    find_cutoff<<<1, BLK, 0, stream>>>(tgt, Pptr, M);
    yolo_main<<<nblocks, BLK, 0, stream>>>(pred, tgt, Pptr, partials, M);
    final_reduce<<<1, BLK, 0, stream>>>(partials, nblocks, out);
}